// PositionalEmbedding_62654982914377
// MI455X (gfx1250) — compile-verified
//
#include <hip/hip_runtime.h>
#include <hip/hip_bf16.h>

// Problem constants (match the JAX reference).
constexpr int BATCH = 8;
constexpr int SEQ   = 4096;
constexpr int DIM   = 1024;

typedef float v4f __attribute__((ext_vector_type(4)));

// angle = s * wave^(-2i/D) = s * exp2(-(2i/D) * log2(wave)).
// All constants are constexpr so they become inline instruction literals
// (no PC-relative .rodata s_load + kmcnt wait at wave start).
constexpr float LOG2_WAVE  = 13.28771237954945f;          // log2(10000)
constexpr float TWO_PI_HI  = 6.28318548202514648438f;     // 0x40C90FDB
constexpr float TWO_PI_LO  = -1.74845552749949e-7f;       // 2*pi - TWO_PI_HI
constexpr float INV_TWO_PI = 0.15915494309189535f;

// One block per sequence position s (4096 blocks, 256 threads = 8 wave32).
// Thread t owns 4 consecutive floats d = 4t..4t+3 == two (sin,cos) pairs.
// The PE pair is computed ONCE per thread and reused for all 8 batches, so
// the transcendental cost is amortized 8x and PE never touches DRAM.
__global__ void __launch_bounds__(256)
PositionalEmbedding_62654982914377_kernel(const float* __restrict__ x,
                                          float* __restrict__ out) {
    const int s  = blockIdx.x;          // 0..SEQ-1
    const int t  = threadIdx.x;         // 0..255
    const int d0 = t << 2;              // first of 4 floats handled here

    const float fs = (float)s;

    // two_i for pair 0 is d0, for pair 1 is d0+2 (even dim index).
    const float invf0 = exp2f(-((float)(d0)     * (1.0f / DIM)) * LOG2_WAVE); // v_exp_f32
    const float invf1 = exp2f(-((float)(d0 + 2) * (1.0f / DIM)) * LOG2_WAVE); // v_exp_f32

    float a0 = fs * invf0;
    float a1 = fs * invf1;

    // Cody-Waite reduction mod 2*pi so hardware v_sin/v_cos stay accurate
    // (max |angle| = 4095 rad; after reduction |r| <= pi).
    float k0 = rintf(a0 * INV_TWO_PI);
    float k1 = rintf(a1 * INV_TWO_PI);
    float r0 = fmaf(-k0, TWO_PI_HI, a0); r0 = fmaf(-k0, TWO_PI_LO, r0);
    float r1 = fmaf(-k1, TWO_PI_HI, a1); r1 = fmaf(-k1, TWO_PI_LO, r1);

    v4f pe;
    pe.x = __sinf(r0);   // v_sin_f32
    pe.y = __cosf(r0);   // v_cos_f32
    pe.z = __sinf(r1);
    pe.w = __cosf(r1);

    const size_t plane = (size_t)SEQ * DIM;              // 4M elements per batch
    const size_t base  = (size_t)s * DIM + (size_t)d0;   // offset within a plane

    // Stream all 8 batches: b128 NT loads/stores, fully unrolled so 8 loads
    // are in flight (one s_clause) before the dependent adds/stores retire.
#pragma unroll
    for (int b = 0; b < BATCH; ++b) {
        const v4f* __restrict__ px = (const v4f*)(x + b * plane + base);
        v4f*       __restrict__ po = (v4f*)(out + b * plane + base);
        v4f v = __builtin_nontemporal_load(px);          // global_load_b128 (NT)
        v.x += pe.x; v.y += pe.y; v.z += pe.z; v.w += pe.w;
        __builtin_nontemporal_store(v, po);              // global_store_b128 (NT)
    }
}

extern "C" void kernel_launch(void* const* d_in, const int* in_sizes, int n_in,
                              void* d_out, int out_size, void* d_ws, size_t ws_size,
                              hipStream_t stream) {
    (void)in_sizes; (void)n_in; (void)d_ws; (void)ws_size; (void)out_size;
    const float* x   = (const float*)d_in[0];
    float*       out = (float*)d_out;

    // DIM/4 = 256 threads per row; one block per sequence position.
    dim3 grid(SEQ);
    dim3 block(DIM / 4);
    PositionalEmbedding_62654982914377_kernel<<<grid, block, 0, stream>>>(x, out);
}